// MultiHeadAttention_75161927680550
// MI455X (gfx1250) — compile-verified
//
#include <hip/hip_runtime.h>

#define N_TOK   4096
#define DMODEL  256
#define NHEADS  8
#define DK      32
#define SCALE   0.17677669529663687f   /* 1/sqrt(32) */

typedef float v2f __attribute__((ext_vector_type(2)));
typedef float v4f __attribute__((ext_vector_type(4)));
typedef float v8f __attribute__((ext_vector_type(8)));

// D = A(16x4 f32) * B(4x16 f32) + C(16x16 f32), wave32, exact fp32 MACs.
__device__ __forceinline__ v8f wmma_f32(v2f a, v2f b, v8f c) {
  return __builtin_amdgcn_wmma_f32_16x16x4_f32(
      /*neg_a=*/false, a, /*neg_b=*/false, b,
      /*c_mod=*/(short)0, c, /*reuse_a=*/false, /*reuse_b=*/false);
}

// ---------------------------------------------------------------------------
// Y[4096,256] = X[4096,256] @ W^T + b      (W is [256,256] row-major)
// One wave per 16x16 output tile; K-loop of 64 f32 WMMAs.
// f32 A fragment layout (ISA 7.12.2): lanes 0-15 hold K={k0,k0+1},
// lanes 16-31 hold K={k0+2,k0+3} for rows M=lane&15.  B mirrors on columns.
// ---------------------------------------------------------------------------
__global__ __launch_bounds__(32)
void proj_kernel(const float* __restrict__ X, const float* __restrict__ W,
                 const float* __restrict__ bias, float* __restrict__ Y)
{
  const int lane = threadIdx.x;
  const int half = lane >> 4;        // 0: K lo pair, 1: K hi pair
  const int l15  = lane & 15;
  const int m0   = (blockIdx.x >> 4) * 16;   // 256 row tiles
  const int n0   = (blockIdx.x & 15) * 16;   // 16  col tiles

  v8f c = {0.f,0.f,0.f,0.f,0.f,0.f,0.f,0.f};
  const float* xrow = X + (m0 + l15) * DMODEL + 2 * half;
  const float* wrow = W + (n0 + l15) * DMODEL + 2 * half;   // B[k][n] = W[n][k]
  for (int k0 = 0; k0 < DMODEL; k0 += 4) {
    v2f a = *(const v2f*)(xrow + k0);
    v2f b = *(const v2f*)(wrow + k0);
    c = wmma_f32(a, b, c);
  }
  const float bv = bias[n0 + l15];
  #pragma unroll
  for (int i = 0; i < 8; ++i)        // C/D: vgpr i -> row i (+8 for hi lanes)
    Y[(m0 + i + 8 * half) * DMODEL + n0 + l15] = c[i] + bv;
}

// ---------------------------------------------------------------------------
// Fused attention: one workgroup per (head, 16-query tile).
// 256 threads = 8 wave32.  LDS holds the full 16x4096 fp32 probability
// row-block (256 KB of the WGP's 320 KB) -> attn hits HBM exactly once,
// as 16-byte non-temporal stores (keeps K/V resident in the 192 MB L2).
// ---------------------------------------------------------------------------
__global__ __launch_bounds__(256)
void attn_kernel(const float* __restrict__ Q, const float* __restrict__ K,
                 const float* __restrict__ V,
                 const int* __restrict__ batch_q, const int* __restrict__ batch_kv,
                 float* __restrict__ attn_out, float* __restrict__ Opre)
{
  extern __shared__ float smem[];
  float* P      = smem;                       // [16][4096] scores -> probs
  float* pv     = smem + 16 * N_TOK;          // [8][16][32] split-K partials
  float* rowred = pv + 8 * 16 * 32;           // [16] row max / row sum
  __shared__ int   qbid[16];
  __shared__ float red[256];

  const int qb   = blockIdx.x;                // 0..255
  const int h    = blockIdx.y;                // 0..7
  const int tid  = threadIdx.x;
  const int lane = tid & 31;
  const int wave = tid >> 5;                  // 0..7 (uniform per wave)
  const int half = lane >> 4;
  const int l15  = lane & 15;
  const int q0   = qb * 16;
  const int hd   = h * DK;

  if (tid < 16) qbid[tid] = batch_q[q0 + tid];
  __syncthreads();

  // --- A fragments of the Q tile (16 x 32), 8 frags covering K=0..31 -------
  v2f afrag[8];
  {
    const float* qrow = Q + (q0 + l15) * DMODEL + hd + 2 * half;
    #pragma unroll
    for (int kk = 0; kk < 8; ++kk) afrag[kk] = *(const v2f*)(qrow + 4 * kk);
  }

  // --- scores: each wave owns 32 of the 256 column tiles -------------------
  for (int t = wave; t < 256; t += 8) {
    const int n0 = t * 16;
    v8f c = {0.f,0.f,0.f,0.f,0.f,0.f,0.f,0.f};
    const float* krow = K + (n0 + l15) * DMODEL + hd + 2 * half; // B[k][n]=K[n][k]
    #pragma unroll
    for (int kk = 0; kk < 8; ++kk) {
      v2f b = *(const v2f*)(krow + 4 * kk);
      c = wmma_f32(afrag[kk], b, c);
    }
    const int col = n0 + l15;
    const int kb  = batch_kv[col];
    #pragma unroll
    for (int i = 0; i < 8; ++i) {
      const int row = i + 8 * half;
      P[row * N_TOK + col] =
          (kb == qbid[row]) ? c[i] * SCALE : -__builtin_inff();
    }
  }
  __syncthreads();

  // --- exact softmax over each 4096-wide row (16 threads/row, b128 LDS) ----
  const int r    = tid >> 4;
  const int subl = tid & 15;
  float m = -__builtin_inff();
  for (int cc = subl * 4; cc < N_TOK; cc += 64) {
    v4f x = *(const v4f*)&P[r * N_TOK + cc];
    m = fmaxf(fmaxf(fmaxf(m, x.x), fmaxf(x.y, x.z)), x.w);
  }
  red[tid] = m;
  __syncthreads();
  if (subl == 0) {
    float mm = red[tid];
    for (int j = 1; j < 16; ++j) mm = fmaxf(mm, red[tid + j]);
    rowred[r] = mm;
  }
  __syncthreads();
  const float rmax = rowred[r];
  float s = 0.f;
  for (int cc = subl * 4; cc < N_TOK; cc += 64) {
    v4f x = *(const v4f*)&P[r * N_TOK + cc];
    x.x = __expf(x.x - rmax);                 // exp(-inf)=0 handles mask
    x.y = __expf(x.y - rmax);
    x.z = __expf(x.z - rmax);
    x.w = __expf(x.w - rmax);
    *(v4f*)&P[r * N_TOK + cc] = x;
    s += (x.x + x.y) + (x.z + x.w);
  }
  red[tid] = s;
  __syncthreads();
  if (subl == 0) {
    float ss = 0.f;
    for (int j = 0; j < 16; ++j) ss += red[tid + j];
    rowred[r] = ss;
  }
  __syncthreads();

  // --- normalize in LDS; stream attn to HBM once: b128 non-temporal --------
  float* attn_h = attn_out + ((size_t)h * N_TOK + q0) * N_TOK;
  for (int rr = 0; rr < 16; ++rr) {
    const float inv = 1.0f / rowred[rr];
    for (int cc = tid * 4; cc < N_TOK; cc += 1024) {
      v4f p = *(const v4f*)&P[rr * N_TOK + cc];
      p.x *= inv; p.y *= inv; p.z *= inv; p.w *= inv;
      *(v4f*)&P[rr * N_TOK + cc] = p;
      __builtin_nontemporal_store(p, (v4f*)&attn_h[(size_t)rr * N_TOK + cc]);
    }
  }
  __syncthreads();

  // --- O_pre tile = P(16x4096) @ V(4096x32), split-K across 8 waves --------
  v8f c0 = {0.f,0.f,0.f,0.f,0.f,0.f,0.f,0.f};
  v8f c1 = {0.f,0.f,0.f,0.f,0.f,0.f,0.f,0.f};
  const int kbeg = wave * (N_TOK / 8);
  for (int k0 = kbeg; k0 < kbeg + (N_TOK / 8); k0 += 4) {
    v2f a = *(const v2f*)(&P[l15 * N_TOK + k0 + 2 * half]);     // A from LDS
    const float* vb = V + (size_t)(k0 + 2 * half) * DMODEL + hd + l15;
    v2f b0; b0.x = vb[0];  b0.y = vb[DMODEL];                   // B[k][n]=V[k][n]
    v2f b1; b1.x = vb[16]; b1.y = vb[DMODEL + 16];
    c0 = wmma_f32(a, b0, c0);
    c1 = wmma_f32(a, b1, c1);
  }
  #pragma unroll
  for (int i = 0; i < 8; ++i) {
    const int row = i + 8 * half;
    pv[(wave * 16 + row) * 32 + l15]      = c0[i];
    pv[(wave * 16 + row) * 32 + 16 + l15] = c1[i];
  }
  __syncthreads();
  for (int e = tid; e < 512; e += 256) {
    const int row = e >> 5, col = e & 31;
    float acc = 0.f;
    #pragma unroll
    for (int w = 0; w < 8; ++w) acc += pv[(w * 16 + row) * 32 + col];
    Opre[(q0 + row) * DMODEL + hd + col] = acc;
  }
}

// ---------------------------------------------------------------------------
extern "C" void kernel_launch(void* const* d_in, const int* in_sizes, int n_in,
                              void* d_out, int out_size, void* d_ws, size_t ws_size,
                              hipStream_t stream) {
  const float* x_q      = (const float*)d_in[0];
  const float* x_kv     = (const float*)d_in[1];
  const int*   batch_q  = (const int*)d_in[2];
  const int*   batch_kv = (const int*)d_in[3];
  const float* Wq = (const float*)d_in[4];
  const float* bq = (const float*)d_in[5];
  const float* Wk = (const float*)d_in[6];
  const float* bk = (const float*)d_in[7];
  const float* Wv = (const float*)d_in[8];
  const float* bv = (const float*)d_in[9];
  const float* Wo = (const float*)d_in[10];
  const float* bo = (const float*)d_in[11];

  const size_t mat = (size_t)N_TOK * DMODEL;        // 1,048,576 floats
  float* Q    = (float*)d_ws;
  float* Kt   = Q + mat;
  float* Vt   = Kt + mat;
  float* Opre = Vt + mat;

  float* out  = (float*)d_out;                       // [4096,256]
  float* attn = out + mat;                           // [8,4096,4096]

  const int projGrid = 256 * 16;                     // 16x16 tiles
  proj_kernel<<<projGrid, 32, 0, stream>>>(x_q,  Wq, bq, Q);
  proj_kernel<<<projGrid, 32, 0, stream>>>(x_kv, Wk, bk, Kt);
  proj_kernel<<<projGrid, 32, 0, stream>>>(x_kv, Wv, bv, Vt);

  const size_t smem = (size_t)(16 * N_TOK + 8 * 16 * 32 + 16) * sizeof(float);
  attn_kernel<<<dim3(256, NHEADS), 256, smem, stream>>>(
      Q, Kt, Vt, batch_q, batch_kv, attn, Opre);

  proj_kernel<<<projGrid, 32, 0, stream>>>(Opre, Wo, bo, out);
}